// RNN_29523605193279
// MI455X (gfx1250) — compile-verified
//
#include <hip/hip_runtime.h>
#include <math.h>

// ---- Problem constants (match reference) ----
#define BATCH 256
#define TLEN  256
#define DW    300   // embedding dim (divisible by 4 -> no K guard in kernel 1)
#define DH    50    // hidden size
#define NCLS  4
#define PRE_STRIDE 50          // pre[65536][50] in workspace (13.1 MB)
#define HSTRIDE 68             // LDS row stride (floats): even (8B-aligned v2f) and
                               // 4*row+col spreads a wave's 64 dwords over 64 banks

typedef __attribute__((ext_vector_type(2))) float v2f;
typedef __attribute__((ext_vector_type(8))) float v8f;

// =====================================================================
// Kernel 1: fused embedding gather + input projection
//   pre[m][n] = sum_k emb[x[m]][k] * W_ih[n][k] + b_ih[n],  m = b*T + t
// One wave per 16-row M tile; 4 x v8f accumulators cover N = 0..63.
// Branch-free inner loop: the N>=50 tail fragment uses a clamped row
// pointer plus a 0/1 lane mask, so the K-loop has no exec divergence.
// Uses V_WMMA_F32_16X16X4_F32 (D = A(16x4) * B(4x16) + C).
// =====================================================================
__global__ __launch_bounds__(128) void embed_proj_kernel(
    const int*   __restrict__ x,
    const float* __restrict__ emb,
    const float* __restrict__ W_ih,
    const float* __restrict__ b_ih,
    float*       __restrict__ pre)
{
    const int lane = threadIdx.x & 31;
    const int wave = threadIdx.x >> 5;
    const int m0   = (blockIdx.x * 4 + wave) * 16;   // first of 16 rows
    const int half = lane >> 4;                      // 0: K={k,k+1}, 1: K={k+2,k+3}
    const int l16  = lane & 15;

    // Each lane's A-row token (row M = l16 of the tile); emb (38.4 MB) is L2-resident.
    const int tok = x[m0 + l16];
    const float* __restrict__ arow = emb + (long)tok * DW;

    // B-fragment row pointers: B[kk][n] = W_ih[n][kk]. Tail tile clamped + masked.
    const int   n3 = 48 + l16;
    const float m3 = (n3 < DH) ? 1.f : 0.f;
    const float* __restrict__ b0p = W_ih + ( 0 + l16) * DW;
    const float* __restrict__ b1p = W_ih + (16 + l16) * DW;
    const float* __restrict__ b2p = W_ih + (32 + l16) * DW;
    const float* __restrict__ b3p = W_ih + ((n3 < DH) ? n3 : (DH - 1)) * DW;

    v8f acc0 = {}; v8f acc1 = {}; v8f acc2 = {}; v8f acc3 = {};

    for (int k = 0; k < DW; k += 4) {
        const int kk = k + half * 2;                 // even -> 8B aligned loads
        v2f a  = *(const v2f*)(arow + kk);           // A[16x4] fragment
        v2f b0 = *(const v2f*)(b0p + kk);
        v2f b1 = *(const v2f*)(b1p + kk);
        v2f b2 = *(const v2f*)(b2p + kk);
        v2f b3 = *(const v2f*)(b3p + kk) * m3;       // zeroed on padded lanes

        acc0 = __builtin_amdgcn_wmma_f32_16x16x4_f32(false, a, false, b0, (short)0, acc0, false, false);
        acc1 = __builtin_amdgcn_wmma_f32_16x16x4_f32(false, a, false, b1, (short)0, acc1, false, false);
        acc2 = __builtin_amdgcn_wmma_f32_16x16x4_f32(false, a, false, b2, (short)0, acc2, false, false);
        acc3 = __builtin_amdgcn_wmma_f32_16x16x4_f32(false, a, false, b3, (short)0, acc3, false, false);
    }

    // Epilogue: C/D layout -> VGPR r holds row M = r + 8*half, col n = nt*16 + l16.
    for (int r = 0; r < 8; ++r) {
        const int m = m0 + r + half * 8;
        float* __restrict__ prow = pre + (long)m * PRE_STRIDE;
        prow[ 0 + l16] = acc0[r] + b_ih[ 0 + l16];
        prow[16 + l16] = acc1[r] + b_ih[16 + l16];
        prow[32 + l16] = acc2[r] + b_ih[32 + l16];
        if (n3 < DH) prow[n3] = acc3[r] + b_ih[n3];
    }
}

// =====================================================================
// Kernel 2: sequential Elman scan + classifier head.
// Block = 16 batch rows (independent across blocks), 4 waves = 4 N-tiles.
// h lives in LDS [16][HSTRIDE]; cols 50..67 stay zero so K-padding is
// free and the 13-step WMMA loop needs no guards. W_hh B-fragments are
// preloaded into registers once (26 VGPRs) and reused for all 256 steps.
// =====================================================================
__global__ __launch_bounds__(128) void rnn_scan_kernel(
    const float* __restrict__ pre,
    const float* __restrict__ W_hh,
    const float* __restrict__ b_hh,
    const float* __restrict__ W_out,
    const float* __restrict__ b_out,
    float*       __restrict__ out)
{
    __shared__ float hLDS[16 * HSTRIDE];

    const int lane = threadIdx.x & 31;
    const int wave = threadIdx.x >> 5;       // N tile 0..3
    const int half = lane >> 4;
    const int l16  = lane & 15;
    const int m0   = blockIdx.x * 16;        // first batch row of this block
    const int n    = wave * 16 + l16;        // output column 0..63

    // h0 = 0; padded cols stay 0 forever.
    for (int i = threadIdx.x; i < 16 * HSTRIDE; i += 128) hLDS[i] = 0.f;
    __syncthreads();

    // Preload W_hh B-fragments for all 13 K-steps: B[kk][n] = W_hh[n][kk].
    // Clamp + mask (one-time cost, keeps the register payload branch-free).
    v2f bfrag[13];
    {
        const float bmask_n = (n < DH) ? 1.f : 0.f;
        const float* __restrict__ brow = W_hh + ((n < DH) ? n : (DH - 1)) * DH;
        for (int ks = 0; ks < 13; ++ks) {
            const int   kk = ks * 4 + half * 2;          // max 50
            const float mk = (kk < DH) ? bmask_n : 0.f;  // pairs stay in-bounds (kk even)
            bfrag[ks] = *(const v2f*)(brow + ((kk < DH) ? kk : 0)) * mk;
        }
    }
    const float bh = (n < DH) ? b_hh[n] : 0.f;

    for (int t = 0; t < TLEN; ++t) {
        v8f acc = {};
        for (int ks = 0; ks < 13; ++ks) {
            const int kk = ks * 4 + half * 2;                   // <= 51; cols >=50 are zero
            v2f a = *(const v2f*)(&hLDS[l16 * HSTRIDE + kk]);   // A = h tile (16x4)
            acc = __builtin_amdgcn_wmma_f32_16x16x4_f32(false, a, false, bfrag[ks],
                                                        (short)0, acc, false, false);
        }
        __syncthreads();                     // all waves done reading old h
        if (n < DH) {
            for (int r = 0; r < 8; ++r) {
                const int mrow = r + half * 8;
                const int b    = m0 + mrow;
                const float p  = pre[((long)b * TLEN + t) * PRE_STRIDE + n];
                hLDS[mrow * HSTRIDE + n] = tanhf(acc[r] + p + bh);
            }
        }
        __syncthreads();                     // new h visible to all waves
    }

    // Classifier head: out[b][c] = h[b] . W_out[c] + b_out[c]
    if (threadIdx.x < 16 * NCLS) {
        const int mrow = threadIdx.x >> 2;
        const int c    = threadIdx.x & 3;
        float s = b_out[c];
        for (int j = 0; j < DH; ++j)
            s += hLDS[mrow * HSTRIDE + j] * W_out[c * DH + j];
        out[(m0 + mrow) * NCLS + c] = s;
    }
}

// =====================================================================
extern "C" void kernel_launch(void* const* d_in, const int* in_sizes, int n_in,
                              void* d_out, int out_size, void* d_ws, size_t ws_size,
                              hipStream_t stream) {
    const int*   x     = (const int*)  d_in[0];
    const float* emb   = (const float*)d_in[1];
    const float* W_ih  = (const float*)d_in[2];
    const float* b_ih  = (const float*)d_in[3];
    const float* W_hh  = (const float*)d_in[4];
    const float* b_hh  = (const float*)d_in[5];
    const float* W_out = (const float*)d_in[6];
    const float* b_out = (const float*)d_in[7];
    float* out = (float*)d_out;
    float* pre = (float*)d_ws;   // 65536 * 50 floats = 13.1 MB scratch

    // 65536 rows / 16 per wave / 4 waves per block = 1024 blocks
    embed_proj_kernel<<<(BATCH * TLEN) / 64, 128, 0, stream>>>(x, emb, W_ih, b_ih, pre);
    // 256 batch rows / 16 per block = 16 blocks
    rnn_scan_kernel<<<BATCH / 16, 128, 0, stream>>>(pre, W_hh, b_hh, W_out, b_out, out);
}